// Camera_83288005804825
// MI455X (gfx1250) — compile-verified
//
#include <hip/hip_runtime.h>
#include <math.h>

// ---------------- problem constants ----------------
#define SZ    384          // IMAGE_SIZE
#define HH    192          // IMAGE_SIZE/2
#define NW    3            // wavelengths
#define ND    16           // depths
#define NM    1024         // MASK_SIZE/2
#define NR    194          // rho_grid length
#define NB    2            // batch
#define NC    3            // channels
#define EPSV  0.001f

typedef float v2f __attribute__((ext_vector_type(2)));
typedef float v8f __attribute__((ext_vector_type(8)));

union V8 { v8f v; float f[8]; };

static __device__ __forceinline__ float2 cmulf(float2 a, float2 b) {
    float2 r; r.x = a.x*b.x - a.y*b.y; r.y = a.x*b.y + a.y*b.x; return r;
}

// =====================================================================
// Kernel 0: init scratch scalars
// =====================================================================
__global__ void k_init(float* sums, unsigned* scaleBits) {
    int t = threadIdx.x;
    if (t < NW*ND) sums[t] = 0.f;
    if (t == 0) *scaleBits = 0u;
}

// =====================================================================
// Kernel 1: psf1d via WMMA.  D = A x B per wavelength:
//   A[d][m] = amp*cos(phase)  (and sin)  -> 16 x 1024
//   B[m][r] = H               -> 1024 x 194 (13 tiles of 16 cols)
// One wave (32 threads) per (wavelength, n-tile); K in steps of 4 using
// V_WMMA_F32_16X16X4_F32.  psf1d = (2pi/(wl*sd))^2 * (re^2 + im^2).
//
// Phase sum is done in f64 (one fma), trig in the f32 TRANS pipe
// (|phase| < ~15 rad, so f64 Payne-Hanek reduction is pure overhead).
// B-column guard is branchless (clamped address * 0/1 mask) so the hot
// loop issues unconditional loads with no exec-mask churn.
// =====================================================================
__global__ __launch_bounds__(32)
void k_psf1d(const double* __restrict__ height,
             const double* __restrict__ amp,
             const double* __restrict__ pph,
             const double* __restrict__ Hmat,
             float* __restrict__ psf1d)
{
    const int w     = blockIdx.x / 13;
    const int n0    = (blockIdx.x % 13) * 16;
    const int lane  = threadIdx.x;
    const int mrow  = lane & 15;   // A row (depth) / B,C column-within-tile
    const int khalf = lane >> 4;

    const double WL[3] = {6.32e-07, 5.5e-07, 4.5e-07};
    const double wl = WL[w];
    const double um2 = (wl*1e6)*(wl*1e6);
    const double nidx = 1.5375 + 0.00829045/um2 - 0.000211046/(um2*um2);
    const double PI_D = 3.14159265358979323846;
    const double phc = 2.0*PI_D/wl*(nidx - 1.0);
    const double SENSOR_DIST = 1.0/(1.0/0.05 - 1.0/1.7);
    const double kf = 2.0*PI_D/(wl*SENSOR_DIST);
    const float  k2 = (float)(kf*kf);

    const double* ampW = amp  + (size_t)w*ND*NM + (size_t)mrow*NM;
    const double* pphW = pph  + (size_t)w*ND*NM + (size_t)mrow*NM;
    const int  ncol  = n0 + mrow;
    const bool valid = (ncol < NR);
    const float bmask = valid ? 1.f : 0.f;               // hoisted 0/1 mask
    const int  ncl   = valid ? ncol : (NR - 1);          // clamped address
    const double* HW = Hmat + (size_t)w*NM*NR + ncl;

    V8 cre = {}; V8 cim = {};
    for (int k0 = 0; k0 < NM; k0 += 4) {
        const int kk = k0 + 2*khalf;           // this lane's K pair: kk, kk+1
        // A operands: row = mrow(depth), cols kk, kk+1
        float p0 = (float)__fma_rn(phc, height[kk],     pphW[kk]);
        float p1 = (float)__fma_rn(phc, height[kk + 1], pphW[kk + 1]);
        float a0 = (float)ampW[kk];
        float a1 = (float)ampW[kk + 1];
        float s0, c0, s1, c1;
        __sincosf(p0, &s0, &c0);
        __sincosf(p1, &s1, &c1);
        v2f are, aim, bb;
        are.x = a0*c0; are.y = a1*c1;
        aim.x = a0*s0; aim.y = a1*s1;
        // B operands: rows kk, kk+1, col = ncol (branchless zero-pad)
        bb.x = bmask * (float)HW[(size_t)kk*NR];
        bb.y = bmask * (float)HW[(size_t)(kk + 1)*NR];
        cre.v = __builtin_amdgcn_wmma_f32_16x16x4_f32(
                    false, are, false, bb, (short)0, cre.v, false, false);
        cim.v = __builtin_amdgcn_wmma_f32_16x16x4_f32(
                    false, aim, false, bb, (short)0, cim.v, false, false);
    }
    // D layout: VGPR j -> (d = j + 8*khalf, r = n0 + mrow)
    if (valid) {
        for (int j = 0; j < 8; ++j) {
            int d = j + 8*khalf;
            psf1d[((size_t)w*ND + d)*NR + ncol] =
                k2*(cre.f[j]*cre.f[j] + cim.f[j]*cim.f[j]);
        }
    }
}

// =====================================================================
// Kernel 2: build un-normalized, mirrored, fftshifted PSF (48 x 384x384)
// via closed-form linear interp on the uniform rho grid, and reduce the
// per-slice sum (folded into spectral multiply later).
// grid: (48, 576) x 256
// =====================================================================
__global__ __launch_bounds__(256)
void k_build_psf(const float* __restrict__ psf1d,
                 const double* __restrict__ rho_grid,
                 const double* __restrict__ rho_samp,
                 float* __restrict__ psfN,
                 float* __restrict__ sums)
{
    const int slice = blockIdx.x;            // w*ND + d
    const int w = slice / ND;
    const int pix = blockIdx.y*blockDim.x + threadIdx.x;   // exactly covers 384*384
    const int y = pix / SZ, x = pix % SZ;
    // undo fftshift, then map into quadrant (mirror)
    const int ys = (y + HH) % SZ, xs = (x + HH) % SZ;
    const int yq = ys < HH ? (HH - 1 - ys) : ys - HH;
    const int xq = xs < HH ? (HH - 1 - xs) : xs - HH;

    const double samp = rho_samp[(size_t)w*HH*HH + (size_t)yq*HH + xq];
    const double g0 = rho_grid[(size_t)w*NR];
    const double rdg = 1.0 / (rho_grid[(size_t)w*NR + 1] - g0);
    float u = (float)((samp - g0) * rdg);
    int i0 = (int)floorf(u);
    i0 = min(max(i0, 0), NR - 2);
    float fr = u - (float)i0;
    fr = fminf(fmaxf(fr, 0.f), 1.f);
    const float* p = psf1d + (size_t)slice*NR;
    float val = p[i0]*(1.f - fr) + p[i0 + 1]*fr;
    val = fmaxf(val, 0.f);
    psfN[(size_t)slice*SZ*SZ + pix] = val;

    __shared__ float red[256];
    red[threadIdx.x] = val;
    __syncthreads();
    for (int s2 = 128; s2 > 0; s2 >>= 1) {
        if (threadIdx.x < s2) red[threadIdx.x] += red[threadIdx.x + s2];
        __syncthreads();
    }
    if (threadIdx.x == 0) atomicAdd(&sums[slice], red[0]);
}

// =====================================================================
// Kernel 3: scale = max(img)  (layering is a partition of unity, so
// volume.max() == img.max()).  Positive floats -> bit-pattern monotonic.
// =====================================================================
__global__ void k_maximg(const float* __restrict__ img, int n,
                         unsigned* __restrict__ scaleBits)
{
    int t = blockIdx.x*blockDim.x + threadIdx.x;
    float m = 0.f;
    for (int i = t; i < n; i += gridDim.x*blockDim.x) m = fmaxf(m, img[i]);
    __shared__ float red[256];
    red[threadIdx.x] = m;
    __syncthreads();
    for (int s2 = 128; s2 > 0; s2 >>= 1) {
        if (threadIdx.x < s2) red[threadIdx.x] = fmaxf(red[threadIdx.x], red[threadIdx.x + s2]);
        __syncthreads();
    }
    if (threadIdx.x == 0) atomicMax(scaleBits, __float_as_uint(red[0]));
}

// =====================================================================
// Kernel 4: layered / reverse-cumsum alpha / scaled volume
// one thread per (b, y, x)
// =====================================================================
__global__ void k_build_vol(const float* __restrict__ img,
                            const float* __restrict__ depth,
                            const unsigned* __restrict__ scaleBits,
                            float* __restrict__ layered,
                            float* __restrict__ cums,
                            float* __restrict__ volume)
{
    int t = blockIdx.x*blockDim.x + threadIdx.x;
    if (t >= NB*SZ*SZ) return;
    const int b = t / (SZ*SZ);
    const int pix = t % (SZ*SZ);
    const float inv_scale = 1.f / __uint_as_float(*scaleBits);
    int idx = (int)floorf(depth[t] * (float)ND);
    idx = min(max(idx, 0), ND - 1);
    for (int d = 0; d < ND; ++d) {
        layered[((size_t)b*ND + d)*SZ*SZ + pix] = (d == idx) ? 1.f : 0.f;
        cums[((size_t)b*ND + d)*SZ*SZ + pix]    = (idx >= d) ? 1.f : 0.f;
    }
    for (int c = 0; c < NC; ++c) {
        float iv = img[((size_t)b*NC + c)*SZ*SZ + pix] * inv_scale;
        for (int d = 0; d < ND; ++d)
            volume[(((size_t)(b*NC + c))*ND + d)*SZ*SZ + pix] = (d == idx) ? iv : 0.f;
    }
}

// =====================================================================
// Kernel 5: batched Stockham FFT, N = 384 = 3 * 2^7 (one radix-3 stage +
// seven radix-2 stages).  Block = (192, 2): two lines per block, one
// butterfly per thread per radix-2 stage.  12 KB LDS.  Handles rows or
// columns, real or complex load/store, forward (-1) or inverse (+1).
// =====================================================================
__global__ __launch_bounds__(384)
void k_fft384(const float* __restrict__ rin, const float2* cin,
              float2* cout, float* __restrict__ rout,
              int colMode, float dirSign, float scale)
{
    const int img  = blockIdx.x;
    const int line = blockIdx.y*2 + threadIdx.y;
    const int tid  = threadIdx.x;
    __shared__ float2 bufA[2][SZ];
    __shared__ float2 bufB[2][SZ];
    float2* x = bufA[threadIdx.y];
    float2* y = bufB[threadIdx.y];
    const size_t base = (size_t)img * SZ * SZ;

    for (int i = tid; i < SZ; i += 192) {
        size_t off = colMode ? base + (size_t)i*SZ + line
                             : base + (size_t)line*SZ + i;
        float2 v;
        if (rin) { v.x = rin[off]; v.y = 0.f; }
        else      v = cin[off];
        x[i] = v;
    }
    __syncthreads();

    const float TWO_PI = 6.2831853071795864769f;

    // ---- radix-3 stage: n = 384, s = 1 ----
    {
        const float th0 = dirSign * TWO_PI / 384.f;
        const float sq  = dirSign * 0.8660254037844386f;   // sin(2pi/3) with sign
        for (int p = tid; p < 128; p += 192) {
            float2 a0 = x[p], a1 = x[p + 128], a2 = x[p + 256];
            float2 t1; t1.x = a1.x + a2.x; t1.y = a1.y + a2.y;
            float2 b0; b0.x = a0.x + t1.x; b0.y = a0.y + t1.y;
            float2 m;  m.x = a0.x - 0.5f*t1.x; m.y = a0.y - 0.5f*t1.y;
            float dx_ = a1.x - a2.x, dy_ = a1.y - a2.y;
            float2 rot; rot.x = -sq*dy_; rot.y = sq*dx_;   // i*s*(a1-a2)
            float2 b1; b1.x = m.x + rot.x; b1.y = m.y + rot.y;
            float2 b2; b2.x = m.x - rot.x; b2.y = m.y - rot.y;
            float c1, s1, c2, s2;
            __sincosf(th0 * (float)p,       &s1, &c1);
            __sincosf(th0 * (float)(2*p),   &s2, &c2);
            y[3*p]     = b0;
            float2 w1; w1.x = c1; w1.y = s1;
            float2 w2; w2.x = c2; w2.y = s2;
            y[3*p + 1] = cmulf(b1, w1);
            y[3*p + 2] = cmulf(b2, w2);
        }
    }
    __syncthreads();
    { float2* t = x; x = y; y = t; }

    // ---- radix-2 stages: n = 128..2, s = 3..192 (m*s == 192 each) ----
    int n = 128, s = 3;
    for (int st = 0; st < 7; ++st) {
        const int m = n >> 1;
        const float thn = dirSign * TWO_PI / (float)n;
        {
            const int p = tid / s;
            const int q = tid - p*s;
            float2 a = x[q + s*p];
            float2 b = x[q + s*(p + m)];
            float2 sum; sum.x = a.x + b.x; sum.y = a.y + b.y;
            float2 dif; dif.x = a.x - b.x; dif.y = a.y - b.y;
            float cw, sw; __sincosf(thn * (float)p, &sw, &cw);
            y[q + s*2*p] = sum;
            float2 r; r.x = dif.x*cw - dif.y*sw; r.y = dif.x*sw + dif.y*cw;
            y[q + s*(2*p + 1)] = r;
        }
        __syncthreads();
        { float2* t = x; x = y; y = t; }
        n = m; s <<= 1;
    }

    // result in x (8 stages -> ends back in bufA)
    for (int i = tid; i < SZ; i += 192) {
        size_t off = colMode ? base + (size_t)i*SZ + line
                             : base + (size_t)line*SZ + i;
        float2 v = x[i];
        if (rout) rout[off] = v.x * scale;
        else { v.x *= scale; v.y *= scale; cout[off] = v; }
    }
}

// =====================================================================
// Kernel 6: spectral multiply with Fpsf (per color,depth) and fold in
// 1/sum(psf) normalization.  expand=1: in indexed (b,d); 0: (b,c,d).
// =====================================================================
__global__ void k_specmul(const float2* __restrict__ specIn,
                          const float2* __restrict__ Fpsf,
                          const float* __restrict__ sums,
                          float2* __restrict__ specOut, int expand)
{
    size_t t = (size_t)blockIdx.x*blockDim.x + threadIdx.x;
    const size_t total = (size_t)NB*NC*ND*SZ*SZ;
    if (t >= total) return;
    size_t pix  = t % ((size_t)SZ*SZ);
    size_t imgi = t / ((size_t)SZ*SZ);
    int d = (int)(imgi % ND);
    int c = (int)((imgi / ND) % NC);
    int b = (int)(imgi / (ND*NC));
    float2 a = expand ? specIn[((size_t)(b*ND + d))*SZ*SZ + pix] : specIn[t];
    float2 f = Fpsf[((size_t)(c*ND + d))*SZ*SZ + pix];
    float inv = 1.f / sums[c*ND + d];
    float2 o = cmulf(a, f);
    o.x *= inv; o.y *= inv;
    specOut[t] = o;
}

// =====================================================================
// Kernel 7: occlusion composite over depth; out = scale * captimg
// =====================================================================
__global__ void k_composite(const float* __restrict__ blurA,
                            const float* __restrict__ blurC,
                            const float* __restrict__ blurV,
                            const unsigned* __restrict__ scaleBits,
                            float* __restrict__ out)
{
    int t = blockIdx.x*blockDim.x + threadIdx.x;
    if (t >= NB*NC*SZ*SZ) return;
    const int pix = t % (SZ*SZ);
    const size_t bc = (size_t)(t / (SZ*SZ));      // b*NC + c
    const size_t base = bc*ND*SZ*SZ + pix;
    float trans = 1.f, acc = 0.f;
    for (int d = 0; d < ND; ++d) {
        size_t off = base + (size_t)d*SZ*SZ;
        float den = blurC[off] + EPSV;
        float bv = blurV[off] / den;
        float ba = blurA[off] / den;
        acc += trans * bv;
        trans *= (1.f - ba);
    }
    out[t] = acc * __uint_as_float(*scaleBits);
}

// =====================================================================
// Host orchestration
// =====================================================================
static inline size_t alignup(size_t x) { return (x + 255) & ~(size_t)255; }

extern "C" void kernel_launch(void* const* d_in, const int* in_sizes, int n_in,
                              void* d_out, int out_size, void* d_ws, size_t ws_size,
                              hipStream_t stream) {
    const float*  img    = (const float*) d_in[0];   // (2,3,384,384) f32
    const float*  depth  = (const float*) d_in[1];   // (2,1,384,384) f32
    const double* height = (const double*)d_in[2];   // (1024,) f64
    const double* amp    = (const double*)d_in[3];   // (3,16,1024) f64
    const double* pph    = (const double*)d_in[4];   // (3,16,1024) f64
    const double* Hmat   = (const double*)d_in[5];   // (3,1024,194) f64
    const double* rgrid  = (const double*)d_in[6];   // (3,194) f64
    const double* rsamp  = (const double*)d_in[7];   // (3,192,192) f64
    float* out = (float*)d_out;                      // (2,3,384,384) f32

    // ---- workspace layout ----
    char* w = (char*)d_ws; size_t off = 0;
    auto take = [&](size_t bytes) { char* p = w + off; off = alignup(off + bytes); return p; };
    const size_t IMGSZ = (size_t)SZ*SZ;
    float*    psf1d   = (float*)   take(sizeof(float)*NW*ND*NR);
    float*    sums    = (float*)   take(sizeof(float)*NW*ND);
    unsigned* scaleB  = (unsigned*)take(sizeof(unsigned));
    float*    psfN    = (float*)   take(sizeof(float)*NW*ND*IMGSZ);
    float*    layered = (float*)   take(sizeof(float)*NB*ND*IMGSZ);
    float*    cums    = (float*)   take(sizeof(float)*NB*ND*IMGSZ);
    float*    volume  = (float*)   take(sizeof(float)*NB*NC*ND*IMGSZ);
    float*    blurA   = (float*)   take(sizeof(float)*NB*NC*ND*IMGSZ);
    float*    blurC   = (float*)   take(sizeof(float)*NB*NC*ND*IMGSZ);
    float*    blurV   = (float*)   take(sizeof(float)*NB*NC*ND*IMGSZ);
    float2*   Fpsf    = (float2*)  take(sizeof(float2)*NW*ND*IMGSZ);
    float2*   specL   = (float2*)  take(sizeof(float2)*NB*ND*IMGSZ);
    float2*   specT   = (float2*)  take(sizeof(float2)*NB*NC*ND*IMGSZ);
    (void)ws_size;

    const dim3 fftBlk(192, 2);

    // 0) init scalars
    k_init<<<1, 64, 0, stream>>>(sums, scaleB);

    // 1) psf1d einsum via WMMA (39 waves)
    k_psf1d<<<NW*13, 32, 0, stream>>>(height, amp, pph, Hmat, psf1d);

    // 2) interp + mirror + fftshift PSF, per-slice sums
    k_build_psf<<<dim3(NW*ND, (unsigned)(IMGSZ/256)), 256, 0, stream>>>(
        psf1d, rgrid, rsamp, psfN, sums);

    // 3) scale = max(img)
    k_maximg<<<1024, 256, 0, stream>>>(img, NB*NC*SZ*SZ, scaleB);

    // 4) layered / cumsum-alpha / scaled volume
    k_build_vol<<<(NB*SZ*SZ + 255)/256, 256, 0, stream>>>(
        img, depth, scaleB, layered, cums, volume);

    // 5) Fpsf = fft2(psfN)
    k_fft384<<<dim3(NW*ND, 192), fftBlk, 0, stream>>>(psfN, nullptr, Fpsf, nullptr, 0, -1.f, 1.f);
    k_fft384<<<dim3(NW*ND, 192), fftBlk, 0, stream>>>(nullptr, Fpsf, Fpsf, nullptr, 1, -1.f, 1.f);

    const size_t totalBCD = (size_t)NB*NC*ND*IMGSZ;
    const unsigned mulGrid = (unsigned)((totalBCD + 255)/256);
    const float invN2 = 1.f/((float)SZ*(float)SZ);

    // 6) blurred_alpha = blur(layered) per color
    k_fft384<<<dim3(NB*ND, 192), fftBlk, 0, stream>>>(layered, nullptr, specL, nullptr, 0, -1.f, 1.f);
    k_fft384<<<dim3(NB*ND, 192), fftBlk, 0, stream>>>(nullptr, specL, specL, nullptr, 1, -1.f, 1.f);
    k_specmul<<<mulGrid, 256, 0, stream>>>(specL, Fpsf, sums, specT, 1);
    k_fft384<<<dim3(NB*NC*ND, 192), fftBlk, 0, stream>>>(nullptr, specT, specT, nullptr, 1, 1.f, 1.f);
    k_fft384<<<dim3(NB*NC*ND, 192), fftBlk, 0, stream>>>(nullptr, specT, nullptr, blurA, 0, 1.f, invN2);

    // 7) blurred_cumsum = blur(cumsum_alpha) per color
    k_fft384<<<dim3(NB*ND, 192), fftBlk, 0, stream>>>(cums, nullptr, specL, nullptr, 0, -1.f, 1.f);
    k_fft384<<<dim3(NB*ND, 192), fftBlk, 0, stream>>>(nullptr, specL, specL, nullptr, 1, -1.f, 1.f);
    k_specmul<<<mulGrid, 256, 0, stream>>>(specL, Fpsf, sums, specT, 1);
    k_fft384<<<dim3(NB*NC*ND, 192), fftBlk, 0, stream>>>(nullptr, specT, specT, nullptr, 1, 1.f, 1.f);
    k_fft384<<<dim3(NB*NC*ND, 192), fftBlk, 0, stream>>>(nullptr, specT, nullptr, blurC, 0, 1.f, invN2);

    // 8) blurred_volume = blur(volume)
    k_fft384<<<dim3(NB*NC*ND, 192), fftBlk, 0, stream>>>(volume, nullptr, specT, nullptr, 0, -1.f, 1.f);
    k_fft384<<<dim3(NB*NC*ND, 192), fftBlk, 0, stream>>>(nullptr, specT, specT, nullptr, 1, -1.f, 1.f);
    k_specmul<<<mulGrid, 256, 0, stream>>>(specT, Fpsf, sums, specT, 0);
    k_fft384<<<dim3(NB*NC*ND, 192), fftBlk, 0, stream>>>(nullptr, specT, specT, nullptr, 1, 1.f, 1.f);
    k_fft384<<<dim3(NB*NC*ND, 192), fftBlk, 0, stream>>>(nullptr, specT, nullptr, blurV, 0, 1.f, invN2);

    // 9) occlusion composite -> out
    k_composite<<<(NB*NC*SZ*SZ + 255)/256, 256, 0, stream>>>(blurA, blurC, blurV, scaleB, out);
}